// TokenClassifierWithCRF_83150566851207
// MI455X (gfx1250) — compile-verified
//
#include <hip/hip_runtime.h>
#include <hip/hip_bf16.h>
#include <stddef.h>

// Problem constants (from reference): B=64, S=512, H=1024, L=64
#define B_  64
#define S_  512
#define H_  1024
#define L_  64
#define M_  (B_ * S_)          // 32768 GEMM rows
#define MT  256                // rows per block tile (8 waves x 2 sixteen-row tiles)
#define KC  64                 // K chunk staged in LDS
#define NCH (H_ / KC)          // 16 K-chunks (even -> clean 2x unroll)

typedef __attribute__((ext_vector_type(2))) float v2f;
typedef __attribute__((ext_vector_type(8))) float v8f;
typedef int v4i_gcc __attribute__((vector_size(16)));   // matches builtin param type

#if __has_builtin(__builtin_amdgcn_wmma_f32_16x16x4_f32)
#define USE_WMMA_F32 1
#endif

#if __has_builtin(__builtin_amdgcn_global_load_async_to_lds_b128) && \
    __has_builtin(__builtin_amdgcn_global_load_async_to_lds_b32)  && \
    __has_builtin(__builtin_amdgcn_s_wait_asynccnt)
#define USE_ASYNC_COPY 1
#endif

// Exact-typed address-space casts for the async-to-LDS builtins
// (arg0: T __device__* (AS1), arg1: T AS(3)*; C-style cast also drops const)
#define G128(p) ((v4i_gcc __attribute__((address_space(1)))*)(p))
#define L128(p) ((v4i_gcc __attribute__((address_space(3)))*)(p))
#define G32(p)  ((int __attribute__((address_space(1)))*)(p))
#define L32(p)  ((int __attribute__((address_space(3)))*)(p))

// Async ops issued per thread per stage: 16 b128 (A) + 16 b32 (W transpose)
#define STAGE_OPS 32

// ---------------------------------------------------------------------------
// Kernel 1: emissions = HS[M,K] * W[K,N] + bias   via fp32 WMMA 16x16x4
// Block: 256 threads (8 waves). Each wave -> 32 rows x 64 cols (2x4 C tiles).
// Double-buffered LDS; next chunk is DMA'd via GLOBAL_LOAD_ASYNC_TO_LDS
// (W transposed by per-lane LDS scatter) while the current chunk computes.
// Chunk loop is unrolled x2 so LDS buffer bases are compile-time constants
// and fold into DS-instruction immediate offsets.
// ---------------------------------------------------------------------------
__global__ __launch_bounds__(256)
void emissions_gemm_kernel(const float* __restrict__ HS,
                           const float* __restrict__ W,
                           const float* __restrict__ bias,
                           float* __restrict__ em)
{
    __shared__ float Alds[2][MT * KC];     // 2 x 64 KB: A tile [row][k]
    __shared__ float WldsT[2][L_ * KC];    // 2 x 16 KB: W chunk transposed [n][k]

    const int tid  = threadIdx.x;
    const int lane = tid & 31;
    const int wave = tid >> 5;               // 0..7
    const int row0 = blockIdx.x * MT;        // block's first M row
    const int wbase = wave * 32;             // wave's 32-row stripe

    // WMMA lane mapping (ISA 7.12.2):
    //  A (16x4 f32): lane holds A[lane%16, 2*(lane/16) + v], v in {0,1}
    //  B (4x16 f32): lane holds B[2*(lane/16) + v, lane%16]
    //  C/D (16x16 f32): VGPR r: lanes 0-15 -> M=r, N=lane; 16-31 -> M=8+r, N=lane-16
    const int mrow  = lane & 15;
    const int koff  = (lane >> 4) * 2;
    const int ncol  = lane & 15;
    const int mbase = (lane >> 4) * 8;

    v8f acc[2][4];
    #pragma unroll
    for (int mt = 0; mt < 2; ++mt)
        #pragma unroll
        for (int nt = 0; nt < 4; ++nt)
            acc[mt][nt] = (v8f){0.f,0.f,0.f,0.f,0.f,0.f,0.f,0.f};

    // ---- stage issue: bring A chunk + transposed W chunk into LDS buffer ----
    auto issue_stage = [&](int buf, int k0) {
#if defined(USE_ASYNC_COPY)
        // A tile: DMA 16 B per lane per op, LDS dest mirrors flat layout.
        #pragma unroll
        for (int i = tid; i < (MT * KC) / 4; i += 256) {
            int r = i / (KC / 4);
            int c = i % (KC / 4);
            __builtin_amdgcn_global_load_async_to_lds_b128(
                G128(HS + (size_t)(row0 + r) * H_ + k0 + c * 4),
                L128(&Alds[buf][i * 4]), 0, 0);
        }
        // W chunk: DMA with per-lane LDS scatter performing the transpose.
        #pragma unroll
        for (int i = tid; i < KC * L_; i += 256) {
            int k = i >> 6;          // i / L_
            int n = i & 63;          // i % L_
            __builtin_amdgcn_global_load_async_to_lds_b32(
                G32(W + (size_t)(k0 + k) * L_ + n),
                L32(&WldsT[buf][n * KC + k]), 0, 0);
        }
#else
        const float4* wg = (const float4*)(W + (size_t)k0 * L_);
        #pragma unroll
        for (int i = tid; i < (KC * L_) / 4; i += 256) {
            float4 v = wg[i];
            int k  = i >> 4;
            int n0 = (i & 15) * 4;
            WldsT[buf][(n0 + 0) * KC + k] = v.x;
            WldsT[buf][(n0 + 1) * KC + k] = v.y;
            WldsT[buf][(n0 + 2) * KC + k] = v.z;
            WldsT[buf][(n0 + 3) * KC + k] = v.w;
        }
        float4* al = (float4*)&Alds[buf][0];
        #pragma unroll
        for (int i = tid; i < (MT * KC) / 4; i += 256) {
            int r = i / (KC / 4);
            int c = i % (KC / 4);
            al[i] = *(const float4*)(HS + (size_t)(row0 + r) * H_ + k0 + c * 4);
        }
#endif
    };

    // ---- compute one staged chunk (LDS bases are compile-time constants) ----
    auto compute_chunk = [&](const float* Ab, const float* Wb) {
#if defined(USE_WMMA_F32)
        #pragma unroll 4
        for (int kk = 0; kk < KC; kk += 4) {
            v2f a0 = *(const v2f*)&Ab[(wbase +      mrow) * KC + kk + koff];
            v2f a1 = *(const v2f*)&Ab[(wbase + 16 + mrow) * KC + kk + koff];
            #pragma unroll
            for (int nt = 0; nt < 4; ++nt) {
                v2f bf = *(const v2f*)&Wb[(nt * 16 + mrow) * KC + kk + koff];
                acc[0][nt] = __builtin_amdgcn_wmma_f32_16x16x4_f32(
                    false, a0, false, bf, (short)0, acc[0][nt], false, false);
                acc[1][nt] = __builtin_amdgcn_wmma_f32_16x16x4_f32(
                    false, a1, false, bf, (short)0, acc[1][nt], false, false);
            }
        }
#else
        for (int kk = 0; kk < KC; ++kk) {
            #pragma unroll
            for (int nt = 0; nt < 4; ++nt) {
                float bv = Wb[(nt * 16 + ncol) * KC + kk];
                #pragma unroll
                for (int mt = 0; mt < 2; ++mt)
                    #pragma unroll
                    for (int r = 0; r < 8; ++r)
                        acc[mt][nt][r] += Ab[(wbase + mt * 16 + mbase + r) * KC + kk] * bv;
            }
        }
#endif
    };

    issue_stage(0, 0);

    for (int c = 0; c < NCH; c += 2) {
        // ---- even chunk: compute buffer 0, prefetch chunk c+1 into buffer 1 ----
        issue_stage(1, (c + 1) * KC);            // c+1 <= NCH-1 always (NCH even)
#if defined(USE_ASYNC_COPY)
        __builtin_amdgcn_s_wait_asynccnt(STAGE_OPS);   // stage c landed
#endif
        __syncthreads();
        compute_chunk(&Alds[0][0], &WldsT[0][0]);
        __syncthreads();

        // ---- odd chunk: compute buffer 1, prefetch chunk c+2 into buffer 0 ----
        if (c + 2 < NCH) {
            issue_stage(0, (c + 2) * KC);
#if defined(USE_ASYNC_COPY)
            __builtin_amdgcn_s_wait_asynccnt(STAGE_OPS);  // stage c+1 landed
#endif
        } else {
#if defined(USE_ASYNC_COPY)
            __builtin_amdgcn_s_wait_asynccnt(0);          // final stage landed
#endif
        }
        __syncthreads();
        compute_chunk(&Alds[1][0], &WldsT[1][0]);
        __syncthreads();
    }

    // ---- Epilogue: add bias, store C tiles ----
    #pragma unroll
    for (int mt = 0; mt < 2; ++mt) {
        #pragma unroll
        for (int nt = 0; nt < 4; ++nt) {
            int col = nt * 16 + ncol;
            float bv = bias[col];
            #pragma unroll
            for (int r = 0; r < 8; ++r) {
                int row = row0 + wbase + mt * 16 + mbase + r;
                em[(size_t)row * L_ + col] = acc[mt][nt][r] + bv;
            }
        }
    }
}

// ---------------------------------------------------------------------------
// Kernel 2: CRF log-likelihood per batch element.
// One block (64 threads = 2 waves) per batch element; thread j owns label j.
// ---------------------------------------------------------------------------
__global__ __launch_bounds__(64)
void crf_llh_kernel(const float* __restrict__ em,
                    const int* __restrict__ mask,
                    const long long* __restrict__ labels,
                    const float* __restrict__ st,
                    const float* __restrict__ et,
                    const float* __restrict__ trans,
                    float* __restrict__ llh)
{
    __shared__ float T[L_ * L_];   // 16 KB transitions
    __shared__ float alpha[L_];
    __shared__ float fred[L_];
    __shared__ int   ired[L_];

    const int b = blockIdx.x;
    const int j = threadIdx.x;

    for (int i = j; i < L_ * L_; i += L_) T[i] = trans[i];

    // ---- sequence end index: sum(mask) - 1 ----
    int cnt = 0;
    for (int t = j; t < S_; t += L_) cnt += mask[b * S_ + t];
    ired[j] = cnt;
    __syncthreads();
    for (int s = 32; s > 0; s >>= 1) { if (j < s) ired[j] += ired[j + s]; __syncthreads(); }
    const int seq_end = ired[0] - 1;
    __syncthreads();

    // ---- numerator: parallel over time steps, reduce ----
    float numpart = 0.f;
    for (int t = 1 + j; t < S_; t += L_) {
        if (mask[b * S_ + t]) {
            int tp = (int)labels[b * S_ + t - 1]; if (tp < 0) tp = 0;  // -100 -> O_ID
            int tc = (int)labels[b * S_ + t];     if (tc < 0) tc = 0;
            numpart += T[tp * L_ + tc] + em[((size_t)b * S_ + t) * L_ + tc];
        }
    }
    if (j == 0) {
        int t0 = (int)labels[b * S_];           if (t0 < 0) t0 = 0;
        int tl = (int)labels[b * S_ + seq_end]; if (tl < 0) tl = 0;
        numpart += st[t0] + em[(size_t)b * S_ * L_ + t0] + et[tl];
    }
    fred[j] = numpart;
    __syncthreads();
    for (int s = 32; s > 0; s >>= 1) { if (j < s) fred[j] += fred[j + s]; __syncthreads(); }
    const float numerator = fred[0];
    __syncthreads();

    // ---- forward algorithm (denominator) ----
    alpha[j] = st[j] + em[(size_t)b * S_ * L_ + j];
    __syncthreads();

    for (int t = 1; t < S_; ++t) {
        float newv = alpha[j];
        if (mask[b * S_ + t]) {
            float mx = -3.4e38f;
            #pragma unroll 8
            for (int i = 0; i < L_; ++i) {
                float v = alpha[i] + T[i * L_ + j];
                mx = fmaxf(mx, v);
            }
            float ssum = 0.f;
            #pragma unroll 8
            for (int i = 0; i < L_; ++i)
                ssum += __expf(alpha[i] + T[i * L_ + j] - mx);
            newv = mx + __logf(ssum) + em[((size_t)b * S_ + t) * L_ + j];
        }
        __syncthreads();
        alpha[j] = newv;
        __syncthreads();
    }

    // ---- log_z = logsumexp_j(alpha[j] + end_trans[j]) ----
    const float v = alpha[j] + et[j];
    fred[j] = v;
    __syncthreads();
    for (int s = 32; s > 0; s >>= 1) { if (j < s) fred[j] = fmaxf(fred[j], fred[j + s]); __syncthreads(); }
    const float mx = fred[0];
    __syncthreads();
    fred[j] = __expf(v - mx);
    __syncthreads();
    for (int s = 32; s > 0; s >>= 1) { if (j < s) fred[j] += fred[j + s]; __syncthreads(); }
    if (j == 0) llh[b] = numerator - (mx + __logf(fred[0]));
}

// ---------------------------------------------------------------------------
// Kernel 3: loss = -mean(llh)
// ---------------------------------------------------------------------------
__global__ __launch_bounds__(64)
void loss_reduce_kernel(const float* __restrict__ llh, float* __restrict__ out)
{
    __shared__ float r[B_];
    const int j = threadIdx.x;
    r[j] = llh[j];
    __syncthreads();
    for (int s = 32; s > 0; s >>= 1) { if (j < s) r[j] += r[j + s]; __syncthreads(); }
    if (j == 0) out[0] = -r[0] / (float)B_;
}

// ---------------------------------------------------------------------------
extern "C" void kernel_launch(void* const* d_in, const int* in_sizes, int n_in,
                              void* d_out, int out_size, void* d_ws, size_t ws_size,
                              hipStream_t stream)
{
    const float*     HS     = (const float*)d_in[0];      // [B,S,H] f32
    const int*       mask   = (const int*)d_in[1];        // [B,S]   i32
    const long long* labels = (const long long*)d_in[2];  // [B,S]   i64
    const float*     W      = (const float*)d_in[3];      // [H,L]   f32
    const float*     bias   = (const float*)d_in[4];      // [L]     f32
    const float*     st     = (const float*)d_in[5];      // [L]     f32
    const float*     et     = (const float*)d_in[6];      // [L]     f32
    const float*     trans  = (const float*)d_in[7];      // [L,L]   f32

    float* out = (float*)d_out;        // [0] = loss, [1..] = emissions [B,S,L]
    float* em  = out + 1;
    float* llh = (float*)d_ws;         // 64 floats scratch

    emissions_gemm_kernel<<<M_ / MT, 256, 0, stream>>>(HS, W, bias, em);
    crf_llh_kernel<<<B_, L_, 0, stream>>>(em, mask, labels, st, et, trans, llh);
    loss_reduce_kernel<<<1, B_, 0, stream>>>(llh, out);
}